// AttnModel_42855183679653
// MI455X (gfx1250) — compile-verified
//
#include <hip/hip_runtime.h>

#define DVEC 512
#define HIST_BINS 4096
#define CAP 4096
#define TOPK 64

typedef __attribute__((ext_vector_type(2))) float v2f;
typedef __attribute__((ext_vector_type(8))) float v8f;

__device__ __forceinline__ unsigned fkey(float x) {
    unsigned u = __float_as_uint(x);
    unsigned mask = (unsigned)(((int)u) >> 31);      // all-ones if negative
    return u ^ (mask | 0x80000000u);                 // monotonic float->uint
}

// ---------------------------------------------------------------- K0: init ws
__global__ void k0_init(unsigned* __restrict__ hist, unsigned* __restrict__ ctr) {
    int i = blockIdx.x * blockDim.x + threadIdx.x;
    if (i < HIST_BINS) hist[i] = 0u;
    if (i == 0) ctr[0] = 0u;
}

// ------------------------------------------------- K1: WMMA GEMV + histogram
// One wave (32 lanes) computes alpha for a 16-row tile using
// v_wmma_f32_16x16x4_f32, accumulating over 128 K-chunks of 4.
// A layout (16x4 f32): lanes 0-15 hold K={0,1}, lanes 16-31 hold K={2,3},
// M = lane % 16.  B (4x16): column 0 carries v[c..c+3] (lane0: K0,K1;
// lane16: K2,K3), all other columns 0 -> D column 0 = dot products.
// B fragments come from LDS with a per-lane base/stride: lane0 -> lv+0 (+4),
// lane16 -> lv+2 (+4), all other lanes -> a zeroed 8B pad at lv+DVEC (+0),
// so each B fragment is one ds_load_b64, no cndmasks.
__global__ void __launch_bounds__(256)
k1_gemv_wmma(const float* __restrict__ v, const float* __restrict__ vs,
             float* __restrict__ alpha, unsigned* __restrict__ hist, int n) {
    __shared__ float lv[DVEC + 2];
    int t = threadIdx.x;
    for (int i = t; i < DVEC / 4; i += blockDim.x)
        ((float4*)lv)[i] = ((const float4*)v)[i];
    if (t < 2) lv[DVEC + t] = 0.0f;                  // zero pad for inactive B lanes
    __syncthreads();

    int wave = t >> 5;
    int lane = t & 31;
    int tile = blockIdx.x * (blockDim.x >> 5) + wave;
    if (tile * 16 >= n) return;                      // wave-uniform exit (EXEC all-1s for WMMA)

    int rrow = tile * 16 + (lane & 15);
    if (rrow >= n) rrow = n - 1;                     // clamp (n is a multiple of 16 here)
    const float* p = vs + (size_t)rrow * DVEC + ((lane >> 4) << 1);
    bool l0 = (lane == 0), l16 = (lane == 16);

    const float* bp = lv + (l0 ? 0 : (l16 ? 2 : DVEC));
    int bstep = (l0 || l16) ? 4 : 0;

    v8f acc0 = {}; v8f acc1 = {};
    #pragma unroll 4
    for (int c = 0; c < DVEC; c += 8) {
        if ((c & 31) == 0)                           // once per 128B line window (folds statically)
            __builtin_prefetch(p + c + 128, 0, 0);   // -> global_prefetch_b8, ~4 lines ahead
        v2f a0 = *(const v2f*)(p + c);
        v2f b0 = *(const v2f*)bp; bp += bstep;
        acc0 = __builtin_amdgcn_wmma_f32_16x16x4_f32(false, a0, false, b0,
                                                     (short)0, acc0, false, false);
        v2f a1 = *(const v2f*)(p + c + 4);
        v2f b1 = *(const v2f*)bp; bp += bstep;
        acc1 = __builtin_amdgcn_wmma_f32_16x16x4_f32(false, a1, false, b1,
                                                     (short)0, acc1, false, false);
    }
    v8f acc = acc0 + acc1;

    // D column 0: lane 0 holds M=0..7 (VGPR 0..7), lane 16 holds M=8..15
    if (l0 || l16) {
        int rbase = tile * 16 + (l16 ? 8 : 0);
        #pragma unroll
        for (int r = 0; r < 8; ++r) {
            int row = rbase + r;
            if (row < n) {
                float x = acc[r];
                alpha[row] = x;
                atomicAdd(&hist[fkey(x) >> 20], 1u);
            }
        }
    }
}

// ------------------------------------------- K2: find threshold bin (1 block)
__global__ void k2_threshold(const unsigned* __restrict__ hist,
                             unsigned* __restrict__ tbin) {
    __shared__ unsigned bins[HIST_BINS];
    __shared__ unsigned part[256];
    int t = threadIdx.x;
    unsigned s = 0;
    for (int j = 0; j < HIST_BINS / 256; ++j) {
        unsigned b = hist[t * (HIST_BINS / 256) + j];
        bins[t * (HIST_BINS / 256) + j] = b;
        s += b;
    }
    part[t] = s;
    __syncthreads();
    if (t == 0) {
        unsigned cum = 0; int bin = 0;
        for (int seg = 255; seg >= 0; --seg) {
            if (cum + part[seg] >= (unsigned)TOPK) {
                for (int j = HIST_BINS / 256 - 1; j >= 0; --j) {
                    int b = seg * (HIST_BINS / 256) + j;
                    cum += bins[b];
                    if (cum >= (unsigned)TOPK) { bin = b; break; }
                }
                break;
            }
            cum += part[seg];
        }
        tbin[0] = (unsigned)bin;
    }
}

// ------------------------------------- K3: collect candidates >= threshold bin
__global__ void k3_collect(const float* __restrict__ alpha,
                           const unsigned* __restrict__ tbin,
                           unsigned* __restrict__ ctr,
                           float* __restrict__ cval, int* __restrict__ cidx, int n) {
    int i = blockIdx.x * blockDim.x + threadIdx.x;
    if (i >= n) return;
    float a = alpha[i];
    if ((fkey(a) >> 20) >= tbin[0]) {
        unsigned pos = atomicAdd(ctr, 1u);
        if (pos < (unsigned)CAP) { cval[pos] = a; cidx[pos] = i; }
    }
}

// -------------------------- K4: exact top-64 + softmax + weighted sum (1 block)
__global__ void __launch_bounds__(256)
k4_topk_softmax(const float* __restrict__ cval, const int* __restrict__ cidx,
                const unsigned* __restrict__ ctr,
                const float* __restrict__ scores, float* __restrict__ out) {
    __shared__ float sv[CAP];
    __shared__ int   si[CAP];
    __shared__ float topv[TOPK];
    __shared__ int   topi[TOPK];
    __shared__ float red[256];
    __shared__ int   redi[256];
    int t = threadIdx.x;
    unsigned c0 = ctr[0];
    int M = (c0 < (unsigned)CAP) ? (int)c0 : CAP;
    for (int i = t; i < M; i += 256) { sv[i] = cval[i]; si[i] = cidx[i]; }
    __syncthreads();

    const float NEGINF = -__builtin_inff();
    for (int k = 0; k < TOPK; ++k) {
        float best = NEGINF; int bi = -1;
        for (int i = t; i < M; i += 256) {
            float x = sv[i];
            if (x > best) { best = x; bi = i; }
        }
        red[t] = best; redi[t] = bi;
        __syncthreads();
        for (int s = 128; s > 0; s >>= 1) {
            if (t < s && red[t + s] > red[t]) { red[t] = red[t + s]; redi[t] = redi[t + s]; }
            __syncthreads();
        }
        if (t == 0) {
            int w = redi[0];
            topv[k] = red[0];
            topi[k] = (w >= 0) ? si[w] : 0;
            if (w >= 0) sv[w] = NEGINF;       // remove winner
        }
        __syncthreads();
    }

    if (t == 0) {
        float m = topv[0];                    // selected in descending order
        float sum = 0.0f, wsum = 0.0f;
        for (int k = 0; k < TOPK; ++k) {
            float e = __expf(topv[k] - m);
            sum += e;
            wsum += e * scores[topi[k]];
        }
        out[0] = wsum / sum;
    }
}

// ---------------------------------------------------------------------- launch
extern "C" void kernel_launch(void* const* d_in, const int* in_sizes, int n_in,
                              void* d_out, int out_size, void* d_ws, size_t ws_size,
                              hipStream_t stream) {
    (void)n_in; (void)out_size; (void)ws_size;
    const float* v      = (const float*)d_in[0];   // [512]
    const float* vs     = (const float*)d_in[1];   // [N, 512]
    const float* scores = (const float*)d_in[2];   // [N]
    int n = in_sizes[2];
    float* out = (float*)d_out;

    char* ws = (char*)d_ws;
    float*    alpha = (float*)ws;
    size_t off = (((size_t)n * sizeof(float)) + 255) & ~(size_t)255;
    unsigned* hist  = (unsigned*)(ws + off); off += (size_t)HIST_BINS * 4;
    unsigned* ctr   = (unsigned*)(ws + off); off += 256;
    unsigned* tbin  = (unsigned*)(ws + off); off += 256;
    float*    cvalp = (float*)(ws + off);    off += (size_t)CAP * 4;
    int*      cidxp = (int*)(ws + off);      off += (size_t)CAP * 4;

    k0_init<<<(HIST_BINS + 255) / 256, 256, 0, stream>>>(hist, ctr);

    int tiles = (n + 15) / 16;
    int wavesPerBlock = 8;                         // 256 threads = 8 wave32
    int blocks = (tiles + wavesPerBlock - 1) / wavesPerBlock;
    k1_gemv_wmma<<<blocks, 256, 0, stream>>>(v, vs, alpha, hist, n);

    k2_threshold<<<1, 256, 0, stream>>>(hist, tbin);
    k3_collect<<<(n + 255) / 256, 256, 0, stream>>>(alpha, tbin, ctr, cvalp, cidxp, n);
    k4_topk_softmax<<<1, 256, 0, stream>>>(cvalp, cidxp, ctr, scores, out);
}